// DTI_Model_42769284333683
// MI455X (gfx1250) — compile-verified
//
#include <hip/hip_runtime.h>
#include <hip/hip_bf16.h>

// ---------------------------------------------------------------------------
// DTI GCN model for MI455X (gfx1250, wave32).
// Dense GEMMs -> v_wmma_f32_16x16x4_f32 (full fp32; GEMMs ~10 GFLOP total,
// not the bottleneck). K specialized at compile time so the inner loop is a
// branch-free unrolled stream of ds_load_b64 / global_load_b32 / v_wmma.
// SpMM aggregation -> wave-per-edge float4 gather + global_atomic_add_f32.
// ---------------------------------------------------------------------------

typedef __attribute__((ext_vector_type(2))) float v2f;
typedef __attribute__((ext_vector_type(8))) float v8f;

#define NC_N 16384
#define NP_N 131072
#define EC_N 131072
#define EP_N 2097152
#define BG   256
#define DF   128

// ---------------------------------------------------------------- utilities
__global__ __launch_bounds__(256) void zero_kernel(float* __restrict__ p, long n) {
    long t = (long)blockIdx.x * blockDim.x + threadIdx.x;
    if (t < n) p[t] = 0.0f;
}

__global__ __launch_bounds__(256) void degree_kernel(const int* __restrict__ src,
                                                     const int* __restrict__ dst,
                                                     float* __restrict__ outdeg,
                                                     float* __restrict__ indeg, int E) {
    int e = blockIdx.x * blockDim.x + threadIdx.x;
    if (e < E) {
        unsafeAtomicAdd(&outdeg[src[e]], 1.0f);
        unsafeAtomicAdd(&indeg[dst[e]], 1.0f);
    }
}

__global__ __launch_bounds__(256) void rsqrt_kernel(float* __restrict__ d, int n) {
    int t = blockIdx.x * blockDim.x + threadIdx.x;
    if (t < n) d[t] = __frsqrt_rn(fmaxf(d[t], 1.0f));
}

__global__ __launch_bounds__(256) void count_kernel(const int* __restrict__ gid,
                                                    float* __restrict__ cnt, int n) {
    int t = blockIdx.x * blockDim.x + threadIdx.x;
    if (t < n) unsafeAtomicAdd(&cnt[gid[t]], 1.0f);
}

// ------------------------------------------------------------ WMMA fp32 GEMM
// Y[n x 128] = (X[n x K] * rowscale) @ W[K x 128]  (+bias, relu optional)
// Block: 256 threads = 8 waves; block handles 16 rows; wave w owns cols
// [16w,16w+16). A tile (16 x K4, row-scaled, zero-padded) staged in LDS.
// K is a template parameter -> fully unrolled, branch-free inner loop.
template <int K, bool SCALE, bool BIAS, bool RELU>
__global__ __launch_bounds__(256) void gemm_wmma_f32(const float* __restrict__ X,
                                                     const float* __restrict__ rowscale,
                                                     const float* __restrict__ W,
                                                     const float* __restrict__ bias,
                                                     float* __restrict__ Y) {
    constexpr int K4 = (K + 3) & ~3;   // K padded to multiple of 4
    constexpr int Kp = K4 + 4;         // +4 row pad (even) vs 64 LDS banks
    __shared__ float As[16 * Kp];

    const int row0 = blockIdx.x * 16;
    const int t    = threadIdx.x;

    // Stage row-scaled A tile; zero-fill the K..K4 padding columns.
    for (int idx = t; idx < 16 * K4; idx += 256) {
        int r = idx / K4, k = idx - r * K4;
        float v = 0.0f;
        if (k < K) {
            v = X[(size_t)(row0 + r) * K + k];
            if (SCALE) v *= rowscale[row0 + r];
        }
        As[r * Kp + k] = v;
    }
    __syncthreads();

    const int wave  = t >> 5;
    const int lane  = t & 31;
    const int col0  = wave * 16;
    const int mrow  = lane & 15;          // A row for this lane
    const int ncol  = lane & 15;          // B/D column for this lane
    const int khalf = (lane >> 4) * 2;    // K sub-offset per lane half (0 or 2)

    const float* __restrict__ Wc = W + col0 + ncol;

    v8f acc = {0.f, 0.f, 0.f, 0.f, 0.f, 0.f, 0.f, 0.f};
#pragma unroll
    for (int k0 = 0; k0 < K4; k0 += 4) {
        const int ka = k0 + khalf;        // even -> 8B-aligned LDS read
        v2f a = *(const v2f*)&As[mrow * Kp + ka];
        int kb0, kb1;
        if constexpr (K4 == K) {          // aligned K: no clamping needed
            kb0 = ka;
            kb1 = ka + 1;
        } else {                          // ragged tail: clamp (A pad is 0,
            kb0 = (ka     < K) ? ka     : K - 1;   //  product contributes 0)
            kb1 = (ka + 1 < K) ? ka + 1 : K - 1;
        }
        v2f b;
        b.x = Wc[(size_t)kb0 * DF];
        b.y = Wc[(size_t)kb1 * DF];
        acc = __builtin_amdgcn_wmma_f32_16x16x4_f32(false, a, false, b,
                                                    (short)0, acc, false, false);
    }

    const int mbase = (lane >> 4) * 8;    // D rows: r + 8*(lane/16)
    const float bv = BIAS ? bias[col0 + ncol] : 0.0f;
#pragma unroll
    for (int r = 0; r < 8; ++r) {
        float v = acc[r] + bv;
        if (RELU) v = fmaxf(v, 0.0f);
        Y[(size_t)(row0 + mbase + r) * DF + col0 + ncol] = v;
    }
}

// ------------------------------------------------------------- SpMM scatter
// One wave per edge: gather 128 floats of h[src] (float4/lane), atomic-add
// into m[dst]. Edge indices are wave-uniform -> scalarized loads.
__global__ __launch_bounds__(256) void scatter_kernel(const float* __restrict__ h,
                                                      const int* __restrict__ src,
                                                      const int* __restrict__ dst,
                                                      float* __restrict__ m, int E) {
    int w = (int)(((long)blockIdx.x * blockDim.x + threadIdx.x) >> 5);
    int lane = threadIdx.x & 31;
    if (w >= E) return;
    int s = src[w], d = dst[w];
    const float4 v = *(const float4*)(h + (size_t)s * DF + lane * 4);
    float* mp = m + (size_t)d * DF + lane * 4;
    unsafeAtomicAdd(mp + 0, v.x);
    unsafeAtomicAdd(mp + 1, v.y);
    unsafeAtomicAdd(mp + 2, v.z);
    unsafeAtomicAdd(mp + 3, v.w);
}

// act = relu(m * nd + b)
__global__ __launch_bounds__(256) void finalize_kernel(const float* __restrict__ m,
                                                       const float* __restrict__ nd,
                                                       const float* __restrict__ bias,
                                                       float* __restrict__ y, int n) {
    long t = (long)blockIdx.x * blockDim.x + threadIdx.x;
    if (t >= (long)n * DF) return;
    int i = (int)(t >> 7), c = (int)(t & 127);
    y[t] = fmaxf(m[t] * nd[i] + bias[c], 0.0f);
}

// pool[gid] += relu(m * nd + b)   (second conv layer: never materialize act)
__global__ __launch_bounds__(256) void finalize_pool_kernel(const float* __restrict__ m,
                                                            const float* __restrict__ nd,
                                                            const float* __restrict__ bias,
                                                            const int* __restrict__ gid,
                                                            float* __restrict__ pool, int n) {
    long t = (long)blockIdx.x * blockDim.x + threadIdx.x;
    if (t >= (long)n * DF) return;
    int i = (int)(t >> 7), c = (int)(t & 127);
    float v = fmaxf(m[t] * nd[i] + bias[c], 0.0f);
    unsafeAtomicAdd(&pool[(size_t)gid[i] * DF + c], v);
}

// combined[b, 0:128] = c_pool/cnt ; combined[b, 128:256] = p_pool/cnt
__global__ __launch_bounds__(256) void combine_kernel(const float* __restrict__ c_pool,
                                                      const float* __restrict__ p_pool,
                                                      const float* __restrict__ c_cnt,
                                                      const float* __restrict__ p_cnt,
                                                      float* __restrict__ combined) {
    int t = blockIdx.x * blockDim.x + threadIdx.x;   // BG*256 threads
    if (t >= BG * 2 * DF) return;
    int b = t >> 8, k = t & 255;
    float v;
    if (k < DF) v = c_pool[b * DF + k] / fmaxf(c_cnt[b], 1.0f);
    else        v = p_pool[b * DF + (k - DF)] / fmaxf(p_cnt[b], 1.0f);
    combined[t] = v;
}

// out[b] = dot(h[b, :128], Wf2) + bf2
__global__ __launch_bounds__(256) void mlp_out_kernel(const float* __restrict__ h,
                                                      const float* __restrict__ Wf2,
                                                      const float* __restrict__ bf2,
                                                      float* __restrict__ out) {
    int b = blockIdx.x * blockDim.x + threadIdx.x;
    if (b >= BG) return;
    float s = bf2[0];
    for (int k = 0; k < DF; ++k) s += h[b * DF + k] * Wf2[k];
    out[b] = s;
}

// ---------------------------------------------------------------------------
extern "C" void kernel_launch(void* const* d_in, const int* in_sizes, int n_in,
                              void* d_out, int out_size, void* d_ws, size_t ws_size,
                              hipStream_t stream) {
    const float* compound_feat = (const float*)d_in[0];
    const float* protein_feat  = (const float*)d_in[1];
    const int*   c_src = (const int*)d_in[2];
    const int*   c_dst = (const int*)d_in[3];
    const int*   p_src = (const int*)d_in[4];
    const int*   p_dst = (const int*)d_in[5];
    const int*   c_gid = (const int*)d_in[6];
    const int*   p_gid = (const int*)d_in[7];
    const float* Wc1 = (const float*)d_in[8];
    const float* bc1 = (const float*)d_in[9];
    const float* Wc2 = (const float*)d_in[10];
    const float* bc2 = (const float*)d_in[11];
    const float* Wp1 = (const float*)d_in[12];
    const float* bp1 = (const float*)d_in[13];
    const float* Wp2 = (const float*)d_in[14];
    const float* bp2 = (const float*)d_in[15];
    const float* Wf1 = (const float*)d_in[16];
    const float* bf1 = (const float*)d_in[17];
    const float* Wf2 = (const float*)d_in[18];
    const float* bf2 = (const float*)d_in[19];
    float* out = (float*)d_out;

    float* ws = (float*)d_ws;
    float* c_od = ws;  ws += NC_N;          // compound rsqrt(out_deg)
    float* c_id = ws;  ws += NC_N;          // compound rsqrt(in_deg)
    float* p_od = ws;  ws += NP_N;
    float* p_id = ws;  ws += NP_N;
    float* bufA = ws;  ws += (size_t)NP_N * DF;
    float* bufB = ws;  ws += (size_t)NP_N * DF;
    float* c_pool = ws; ws += BG * DF;
    float* p_pool = ws; ws += BG * DF;
    float* c_cnt  = ws; ws += BG;
    float* p_cnt  = ws; ws += BG;
    float* combined = ws; ws += BG * 2 * DF;
    float* mlph     = ws; ws += BG * DF;

    const int T = 256;
    auto blocks = [](long n) { return (unsigned)((n + 255) / 256); };

    // degrees / counts / pools zeroed (d_ws is poisoned with 0xAA)
    zero_kernel<<<blocks(2L * NC_N), T, 0, stream>>>(c_od, 2L * NC_N);
    zero_kernel<<<blocks(2L * NP_N), T, 0, stream>>>(p_od, 2L * NP_N);
    zero_kernel<<<blocks(2L * BG * DF + 2 * BG), T, 0, stream>>>(c_pool,
                                                                 2L * BG * DF + 2 * BG);

    degree_kernel<<<blocks(EC_N), T, 0, stream>>>(c_src, c_dst, c_od, c_id, EC_N);
    degree_kernel<<<blocks(EP_N), T, 0, stream>>>(p_src, p_dst, p_od, p_id, EP_N);
    rsqrt_kernel<<<blocks(NC_N), T, 0, stream>>>(c_od, NC_N);
    rsqrt_kernel<<<blocks(NC_N), T, 0, stream>>>(c_id, NC_N);
    rsqrt_kernel<<<blocks(NP_N), T, 0, stream>>>(p_od, NP_N);
    rsqrt_kernel<<<blocks(NP_N), T, 0, stream>>>(p_id, NP_N);
    count_kernel<<<blocks(NC_N), T, 0, stream>>>(c_gid, c_cnt, NC_N);
    count_kernel<<<blocks(NP_N), T, 0, stream>>>(p_gid, p_cnt, NP_N);

    // ---------------- compound branch ----------------
    gemm_wmma_f32<74, true, false, false><<<NC_N / 16, T, 0, stream>>>(
        compound_feat, c_od, Wc1, nullptr, bufA);
    zero_kernel<<<blocks((long)NC_N * DF), T, 0, stream>>>(bufB, (long)NC_N * DF);
    scatter_kernel<<<blocks((long)EC_N * 32), T, 0, stream>>>(bufA, c_src, c_dst,
                                                              bufB, EC_N);
    finalize_kernel<<<blocks((long)NC_N * DF), T, 0, stream>>>(bufB, c_id, bc1,
                                                               bufA, NC_N);
    gemm_wmma_f32<128, true, false, false><<<NC_N / 16, T, 0, stream>>>(
        bufA, c_od, Wc2, nullptr, bufB);
    zero_kernel<<<blocks((long)NC_N * DF), T, 0, stream>>>(bufA, (long)NC_N * DF);
    scatter_kernel<<<blocks((long)EC_N * 32), T, 0, stream>>>(bufB, c_src, c_dst,
                                                              bufA, EC_N);
    finalize_pool_kernel<<<blocks((long)NC_N * DF), T, 0, stream>>>(bufA, c_id, bc2,
                                                                    c_gid, c_pool, NC_N);

    // ---------------- protein branch ----------------
    gemm_wmma_f32<128, true, false, false><<<NP_N / 16, T, 0, stream>>>(
        protein_feat, p_od, Wp1, nullptr, bufA);
    zero_kernel<<<blocks((long)NP_N * DF), T, 0, stream>>>(bufB, (long)NP_N * DF);
    scatter_kernel<<<blocks((long)EP_N * 32), T, 0, stream>>>(bufA, p_src, p_dst,
                                                              bufB, EP_N);
    finalize_kernel<<<blocks((long)NP_N * DF), T, 0, stream>>>(bufB, p_id, bp1,
                                                               bufA, NP_N);
    gemm_wmma_f32<128, true, false, false><<<NP_N / 16, T, 0, stream>>>(
        bufA, p_od, Wp2, nullptr, bufB);
    zero_kernel<<<blocks((long)NP_N * DF), T, 0, stream>>>(bufA, (long)NP_N * DF);
    scatter_kernel<<<blocks((long)EP_N * 32), T, 0, stream>>>(bufB, p_src, p_dst,
                                                              bufA, EP_N);
    finalize_pool_kernel<<<blocks((long)NP_N * DF), T, 0, stream>>>(bufA, p_id, bp2,
                                                                    p_gid, p_pool, NP_N);

    // ---------------- readout MLP ----------------
    combine_kernel<<<blocks(BG * 2 * DF), T, 0, stream>>>(c_pool, p_pool, c_cnt,
                                                          p_cnt, combined);
    gemm_wmma_f32<256, false, true, true><<<BG / 16, T, 0, stream>>>(
        combined, nullptr, Wf1, bf1, mlph);
    mlp_out_kernel<<<1, T, 0, stream>>>(mlph, Wf2, bf2, out);
}